// PaDiM2_export_16269336117886
// MI455X (gfx1250) — compile-verified
//
#include <hip/hip_runtime.h>
#include <math.h>

#define BATCH 32
#define DCH   200
#define H     56
#define W     56
#define HW    (H*W)        // 3136
#define C1    64
#define C2    128
#define C3    256
#define PIX_PER_BLK 8
#define KC    40           // K-chunk (5 chunks cover K=200 exactly)
#define LDS_STRIDE 324     // 40*8 + 4 pad: 16B-aligned, spreads banks

typedef float v2f __attribute__((ext_vector_type(2)));
typedef float v8f __attribute__((ext_vector_type(8)));

// ---------------------------------------------------------------------------
// Kernel 1: NN-upsample + concat + channel gather + mean subtract
//   x[p][j][b] = fmaps_concat[b, sel[j], h, w] - mean[j, h, w]
// ---------------------------------------------------------------------------
__global__ __launch_bounds__(256) void k_build_x(
    const float* __restrict__ f1, const float* __restrict__ f2,
    const float* __restrict__ f3, const float* __restrict__ mean,
    const int* __restrict__ sel, float* __restrict__ xbuf)
{
    int idx = blockIdx.x * 256 + threadIdx.x;           // [0, 3136*200*32)
    int b = idx & (BATCH - 1);
    int t = idx >> 5;
    int j = t % DCH;
    int p = t / DCH;
    int h = p / W;
    int w = p - h * W;
    int c = sel[j];
    float v;
    if (c < C1) {
        v = f1[((b * C1 + c) * H + h) * W + w];
    } else if (c < C1 + C2) {
        v = f2[((b * C2 + (c - C1)) * (H/2) + (h >> 1)) * (W/2) + (w >> 1)];
    } else {
        v = f3[((b * C3 + (c - C1 - C2)) * (H/4) + (h >> 2)) * (W/4) + (w >> 2)];
    }
    xbuf[(size_t)p * (DCH * BATCH) + j * BATCH + b] = v - mean[j * HW + p];
}

// ---------------------------------------------------------------------------
// Kernel 2: per-pixel GEMM  left = cov_p (200x200) @ x_p (200x32)  via
// v_wmma_f32_16x16x4_f32, fused with  s[b] = sqrt(sum_i x[i,b]*left[i,b]).
// 1 wave == 1 pixel; 8 waves/block; cov chunks staged via LDS (float4 loads).
// ---------------------------------------------------------------------------
__global__ __launch_bounds__(256) void k_pixel_gemm(
    const float* __restrict__ cov, const float* __restrict__ xbuf,
    float* __restrict__ smap)
{
    __shared__ float lds[16 * LDS_STRIDE];

    const int tid  = threadIdx.x;
    const int lane = tid & 31;
    const int lo   = lane & 15;        // N (or M) position within tile
    const int hi   = lane >> 4;        // K-half selector for A/B fragments
    const int wv   = tid >> 5;         // wave id == pixel within block
    const int p0   = blockIdx.x * PIX_PER_BLK;
    const int p    = p0 + wv;
    const float* __restrict__ xp = xbuf + (size_t)p * (DCH * BATCH);

    float sacc0 = 0.f, sacc1 = 0.f;    // partial s for b=lo and b=16+lo

    for (int mt = 0; mt < 13; ++mt) {              // 13 M-tiles cover 200 (pad 208)
        const int m0 = mt * 16;
        v8f acc0 = {0.f,0.f,0.f,0.f,0.f,0.f,0.f,0.f};
        v8f acc1 = {0.f,0.f,0.f,0.f,0.f,0.f,0.f,0.f};

        for (int kc = 0; kc < 5; ++kc) {           // K chunks of 40
            const int k0 = kc * KC;
            __syncthreads();
            // cooperative stage: cov[m0:m0+16, k0:k0+40, p0:p0+8] -> LDS
            for (int u = tid; u < 16 * KC * 2; u += 256) {
                int mi   = u / (KC * 2);
                int rem  = u - mi * (KC * 2);
                int kj   = rem >> 1;
                int half = rem & 1;
                int mrow = m0 + mi; mrow = (mrow > DCH - 1) ? (DCH - 1) : mrow;
                const float4 g = *(const float4*)(
                    cov + (size_t)(mrow * DCH + k0 + kj) * HW + p0 + 4 * half);
                *(float4*)(&lds[mi * LDS_STRIDE + kj * 8 + 4 * half]) = g;
            }
            __syncthreads();

            #pragma unroll
            for (int ks = 0; ks < KC / 4; ++ks) {  // 10 wmma k-steps of 4
                const int kk = ks * 4;
                // A fragment (16x4 f32): lane=m (lo), vgpr0: k=kk+2*hi, vgpr1: +1
                v2f a;
                const int abase = lo * LDS_STRIDE + (kk + 2 * hi) * 8 + wv;
                a.x = lds[abase];
                a.y = lds[abase + 8];
                // B fragments (4x16 f32): lane=n (lo), k = k0+kk+2*hi (+1)
                const int kB = k0 + kk + 2 * hi;
                v2f b0, b1;
                b0.x = xp[kB * BATCH + lo];
                b0.y = xp[(kB + 1) * BATCH + lo];
                b1.x = xp[kB * BATCH + 16 + lo];
                b1.y = xp[(kB + 1) * BATCH + 16 + lo];
                acc0 = __builtin_amdgcn_wmma_f32_16x16x4_f32(
                    false, a, false, b0, (short)0, acc0, false, false);
                acc1 = __builtin_amdgcn_wmma_f32_16x16x4_f32(
                    false, a, false, b1, (short)0, acc1, false, false);
            }
        }

        // fold left-tile into s:  s[b] += x[m,b] * left[m,b]
        // D layout: vgpr r holds m = m0 + r + 8*hi, column n = lo
        #pragma unroll
        for (int r = 0; r < 8; ++r) {
            int m  = m0 + r + 8 * hi;
            int mc = (m > DCH - 1) ? (DCH - 1) : m;
            float valid = (m < DCH) ? 1.f : 0.f;
            float xv0 = xp[mc * BATCH + lo];
            float xv1 = xp[mc * BATCH + 16 + lo];
            sacc0 += valid * acc0[r] * xv0;
            sacc1 += valid * acc1[r] * xv1;
        }
    }

    // combine the two lane-halves (m 0-7 | 8-15 of each tile) across the wave
    sacc0 += __shfl_xor(sacc0, 16, 32);
    sacc1 += __shfl_xor(sacc1, 16, 32);
    float s = (lane < 16) ? sacc0 : sacc1;   // lane<16 -> b=lane; else b=lane
    smap[p * BATCH + lane] = sqrtf(fmaxf(s, 0.f));
}

// ---------------------------------------------------------------------------
// Kernel 3: bilinear x4 upsample (half-pixel centers, edge clamp) + normalize
// ---------------------------------------------------------------------------
__global__ __launch_bounds__(256) void k_bilinear(
    const float* __restrict__ smap, const float* __restrict__ minp,
    const float* __restrict__ maxp, float* __restrict__ out)
{
    int idx = blockIdx.x * 256 + threadIdx.x;
    if (idx >= BATCH * 224 * 224) return;
    int ow = idx % 224;
    int oh = (idx / 224) % 224;
    int b  = idx / (224 * 224);

    float fy = (oh + 0.5f) * 0.25f - 0.5f;
    float fx = (ow + 0.5f) * 0.25f - 0.5f;
    int y0 = (int)floorf(fy); float wy = fy - (float)y0;
    int x0 = (int)floorf(fx); float wx = fx - (float)x0;
    int y1 = y0 + 1, x1 = x0 + 1;
    y0 = min(max(y0, 0), H - 1); y1 = min(max(y1, 0), H - 1);
    x0 = min(max(x0, 0), W - 1); x1 = min(max(x1, 0), W - 1);

    float v00 = smap[(y0 * W + x0) * BATCH + b];
    float v01 = smap[(y0 * W + x1) * BATCH + b];
    float v10 = smap[(y1 * W + x0) * BATCH + b];
    float v11 = smap[(y1 * W + x1) * BATCH + b];
    float v = (1.f - wy) * ((1.f - wx) * v00 + wx * v01)
            +        wy  * ((1.f - wx) * v10 + wx * v11);

    float mn = *minp, mx = *maxp;
    out[idx] = (v - mn) / (mx - mn);
}

// ---------------------------------------------------------------------------
extern "C" void kernel_launch(void* const* d_in, const int* in_sizes, int n_in,
                              void* d_out, int out_size, void* d_ws, size_t ws_size,
                              hipStream_t stream) {
    const float* fmap1 = (const float*)d_in[0];
    const float* fmap2 = (const float*)d_in[1];
    const float* fmap3 = (const float*)d_in[2];
    const float* mean  = (const float*)d_in[3];
    const float* cov   = (const float*)d_in[4];
    const float* minp  = (const float*)d_in[5];
    const float* maxp  = (const float*)d_in[6];
    const int*   sel   = (const int*)d_in[7];
    float* out  = (float*)d_out;

    float* xbuf = (float*)d_ws;                               // 3136*200*32 f32
    float* smap = xbuf + (size_t)HW * DCH * BATCH;            // 3136*32 f32

    k_build_x<<<(HW * DCH * BATCH) / 256, 256, 0, stream>>>(
        fmap1, fmap2, fmap3, mean, sel, xbuf);
    k_pixel_gemm<<<HW / PIX_PER_BLK, 256, 0, stream>>>(cov, xbuf, smap);
    k_bilinear<<<(BATCH * 224 * 224 + 255) / 256, 256, 0, stream>>>(
        smap, minp, maxp, out);
}